// GCNWithReg_37426345018067
// MI455X (gfx1250) — compile-verified
//
#include <hip/hip_runtime.h>
#include <math.h>

#define NN 40000
#define NE 640000
#define DOUT 40
#define HSIZE (1u << 21)
#define HMASK (HSIZE - 1u)

typedef __attribute__((ext_vector_type(2))) float v2f;
typedef __attribute__((ext_vector_type(8))) float v8f;

__device__ __forceinline__ void atomic_add_f32(float* p, float v) {
    __hip_atomic_fetch_add(p, v, __ATOMIC_RELAXED, __HIP_MEMORY_SCOPE_AGENT);
}

__global__ __launch_bounds__(256) void fill_u32_kernel(unsigned* __restrict__ p, unsigned v, int n) {
    int i = blockIdx.x * 256 + threadIdx.x;
    if (i < n) p[i] = v;
}

__global__ __launch_bounds__(256) void deg_kernel(const int* __restrict__ dst,
                                                  float* __restrict__ deg, int E) {
    int e = blockIdx.x * 256 + threadIdx.x;
    if (e < E) atomic_add_f32(&deg[dst[e]], 1.0f);
}

__global__ __launch_bounds__(256) void rsqrt_kernel(float* __restrict__ dis, int n) {
    int i = blockIdx.x * 256 + threadIdx.x;
    if (i < n) dis[i] = rsqrtf(dis[i] + 1.0f);  // +1 for the self-loop
}

__global__ __launch_bounds__(256) void padW3_kernel(const float* __restrict__ W3,
                                                    float* __restrict__ Wp) {
    int i = blockIdx.x * 256 + threadIdx.x;  // 128*48 elements
    if (i < 128 * 48) {
        int k = i / 48, c = i % 48;
        Wp[i] = (c < DOUT) ? W3[k * DOUT + c] : 0.0f;
    }
}

// Fused GEMM + GCN epilogue.
//   H[r][c]   = (RELU ? relu(A) : A) @ B            (messages for the edge scatter)
//   AGG[r][c] = bias[c] + dis[r]^2 * H[r][c]        (self-loop term + bias, scatter target)
// One wave per 16x16 tile via V_WMMA_F32_16X16X4_F32; 8 waves/block cover N<=128.
// f32 WMMA frags: lanes 0-15 hold K=k0,k0+1; lanes 16-31 hold K=k0+2,k0+3 (A and B);
// C/D: VGPR v -> row v (lanes 0-15) / row v+8 (lanes 16-31), N = lane&15.
template <bool RELU>
__global__ __launch_bounds__(256) void gemm_fused_kernel(const float* __restrict__ A,
                                                         const float* __restrict__ B,
                                                         const float* __restrict__ bias,
                                                         const float* __restrict__ dis,
                                                         float* __restrict__ H,
                                                         float* __restrict__ AGG,
                                                         int ldb, int ncols, int ldc) {
    const int lane = threadIdx.x & 31;
    const int wave = threadIdx.x >> 5;
    const int n0 = wave * 16;
    if (n0 >= ldb) return;  // wave-uniform; inactive tiles for the narrow layer
    const int half = lane >> 4;
    const int m = lane & 15;
    const int row = blockIdx.x * 16 + m;
    const float* Arow = A + (size_t)row * 128;
    const int nb = n0 + m;  // N index for B loads and stores
    v8f acc = {};
#pragma unroll 4
    for (int k0 = 0; k0 < 128; k0 += 4) {
        const int ka = k0 + half * 2;
        v2f a;
        a.x = Arow[ka];
        a.y = Arow[ka + 1];
        if (RELU) {
            a.x = fmaxf(a.x, 0.0f);
            a.y = fmaxf(a.y, 0.0f);
        }
        v2f b;
        b.x = B[ka * ldb + nb];
        b.y = B[(ka + 1) * ldb + nb];
        acc = __builtin_amdgcn_wmma_f32_16x16x4_f32(false, a, false, b, (short)0, acc,
                                                    false, false);
    }
    if (nb < ncols) {
        float bv = bias[nb];
#pragma unroll
        for (int v = 0; v < 8; ++v) {
            int cm = blockIdx.x * 16 + v + half * 8;
            float val = acc[v];
            size_t idx = (size_t)cm * ldc + nb;
            H[idx] = val;
            float s = dis[cm];
            AGG[idx] = bv + s * s * val;
        }
    }
}

// one wave per edge, float4 per lane over 128 channels
__global__ __launch_bounds__(256) void scatter128_kernel(const float* __restrict__ h,
                                                         const int* __restrict__ src,
                                                         const int* __restrict__ dst,
                                                         const float* __restrict__ dis,
                                                         float* __restrict__ out, int E) {
    int e = blockIdx.x * 8 + (threadIdx.x >> 5);
    if (e >= E) return;
    int lane = threadIdx.x & 31;
    int s = src[e], d = dst[e];
    float norm = dis[s] * dis[d];
    const float4* hv = (const float4*)(h + (size_t)s * 128);
    float4 v = hv[lane];
    float* o = out + (size_t)d * 128 + lane * 4;
    atomic_add_f32(o + 0, norm * v.x);
    atomic_add_f32(o + 1, norm * v.y);
    atomic_add_f32(o + 2, norm * v.z);
    atomic_add_f32(o + 3, norm * v.w);
}

// one wave per edge over 40 channels
__global__ __launch_bounds__(256) void scatter40_kernel(const float* __restrict__ h,
                                                        const int* __restrict__ src,
                                                        const int* __restrict__ dst,
                                                        const float* __restrict__ dis,
                                                        float* __restrict__ out, int E) {
    int e = blockIdx.x * 8 + (threadIdx.x >> 5);
    if (e >= E) return;
    int lane = threadIdx.x & 31;
    int s = src[e], d = dst[e];
    float norm = dis[s] * dis[d];
    const float* hr = h + (size_t)s * 40;
    float* o = out + (size_t)d * 40;
    atomic_add_f32(o + lane, norm * hr[lane]);
    if (lane < 8) atomic_add_f32(o + 32 + lane, norm * hr[32 + lane]);
}

__device__ __forceinline__ unsigned hkey(int key) {
    unsigned h = (unsigned)key * 2654435761u;
    h ^= h >> 15;
    return h & HMASK;
}

__global__ __launch_bounds__(256) void hash_insert_kernel(const int* __restrict__ src,
                                                          const int* __restrict__ dst,
                                                          int* __restrict__ keys,
                                                          int* __restrict__ counts, int E, int n) {
    int e = blockIdx.x * 256 + threadIdx.x;
    if (e >= E) return;
    int key = src[e] * n + dst[e];
    unsigned slot = hkey(key);
    while (true) {
        int prev = atomicCAS(&keys[slot], -1, key);
        if (prev == -1 || prev == key) {
            atomicAdd(&counts[slot], 1);
            break;
        }
        slot = (slot + 1) & HMASK;
    }
}

__global__ __launch_bounds__(256) void hash_lookup_kernel(const int* __restrict__ src,
                                                          const int* __restrict__ dst,
                                                          const int* __restrict__ keys,
                                                          const int* __restrict__ counts,
                                                          int* __restrict__ accum, int E, int n) {
    int e = blockIdx.x * 256 + threadIdx.x;
    if (e >= E) return;
    int rev = dst[e] * n + src[e];
    unsigned slot = hkey(rev);
    int cnt = 0;
    while (true) {
        int k = keys[slot];
        if (k == -1) break;
        if (k == rev) {
            cnt = counts[slot];
            break;
        }
        slot = (slot + 1) & HMASK;
    }
    if (cnt) atomicAdd(accum, cnt);
}

__global__ void finalize_reg_kernel(const int* __restrict__ accum, float* __restrict__ out) {
    if (threadIdx.x == 0 && blockIdx.x == 0) out[0] = (float)(*accum);
}

// in-place log_softmax over rows of 40; one wave per row
__global__ __launch_bounds__(256) void log_softmax_kernel(float* __restrict__ out, int n) {
    int row = blockIdx.x * 8 + (threadIdx.x >> 5);
    if (row >= n) return;
    int lane = threadIdx.x & 31;
    float* r = out + (size_t)row * 40;
    float x0 = r[lane];
    bool has1 = (lane < 8);
    float x1 = has1 ? r[lane + 32] : -INFINITY;
    float m = fmaxf(x0, x1);
    for (int o = 16; o > 0; o >>= 1) m = fmaxf(m, __shfl_xor(m, o, 32));
    float s = expf(x0 - m) + (has1 ? expf(x1 - m) : 0.0f);
    for (int o = 16; o > 0; o >>= 1) s += __shfl_xor(s, o, 32);
    float l = logf(s);
    r[lane] = x0 - m - l;
    if (has1) r[lane + 32] = x1 - m - l;
}

extern "C" void kernel_launch(void* const* d_in, const int* in_sizes, int n_in,
                              void* d_out, int out_size, void* d_ws, size_t ws_size,
                              hipStream_t stream) {
    const float* x = (const float*)d_in[0];
    const int* ei = (const int*)d_in[1];
    const float* W1 = (const float*)d_in[2];
    const float* b1 = (const float*)d_in[3];
    const float* W2 = (const float*)d_in[4];
    const float* b2 = (const float*)d_in[5];
    const float* W3 = (const float*)d_in[6];
    const float* b3 = (const float*)d_in[7];
    const int n = NN, E = NE;
    const int* src = ei;
    const int* dst = ei + E;

    char* ws = (char*)d_ws;
    float* dis = (float*)(ws);                   // 40000 f32
    float* bufA = (float*)(ws + 160000);         // 40000*128  (h messages)
    float* bufB = (float*)(ws + 20640000);       // 40000*128  (layer-1 agg)
    float* bufD = (float*)(ws + 41120000);       // 40000*128  (layer-2 agg)
    float* bufC = (float*)(ws + 61600000);       // 40000*40   (layer-3 h)
    float* W3p = (float*)(ws + 68000000);        // 128*48 padded W3
    int* hkeys = (int*)(ws + 68024576);          // 2M slots
    int* hcnts = hkeys + HSIZE;
    int* accum = hcnts + HSIZE;
    float* outp = (float*)d_out;

    dim3 b256(256);
    // ---- init scratch (fill kernels: graph-capture safe) ----
    fill_u32_kernel<<<(n + 255) / 256, b256, 0, stream>>>((unsigned*)dis, 0u, n);
    fill_u32_kernel<<<((int)HSIZE + 255) / 256, b256, 0, stream>>>((unsigned*)hkeys, 0xFFFFFFFFu,
                                                                   (int)HSIZE);
    fill_u32_kernel<<<((int)HSIZE + 256) / 256, b256, 0, stream>>>((unsigned*)hcnts, 0u,
                                                                   (int)HSIZE + 1);  // + accum
    // ---- degree / normalization ----
    deg_kernel<<<(E + 255) / 256, b256, 0, stream>>>(dst, dis, E);
    rsqrt_kernel<<<(n + 255) / 256, b256, 0, stream>>>(dis, n);
    padW3_kernel<<<(128 * 48 + 255) / 256, b256, 0, stream>>>(W3, W3p);
    // ---- reg_loss = trace(A@A) via hash multiset ----
    hash_insert_kernel<<<(E + 255) / 256, b256, 0, stream>>>(src, dst, hkeys, hcnts, E, n);
    hash_lookup_kernel<<<(E + 255) / 256, b256, 0, stream>>>(src, dst, hkeys, hcnts, accum, E, n);
    finalize_reg_kernel<<<1, 1, 0, stream>>>(accum, outp + (size_t)n * 40);
    // ---- layer 1: h1 = x@W1 -> bufA; bufB = b1 + dis^2*h1; scatter into bufB ----
    gemm_fused_kernel<false><<<n / 16, b256, 0, stream>>>(x, W1, b1, dis, bufA, bufB, 128, 128,
                                                          128);
    scatter128_kernel<<<(E + 7) / 8, b256, 0, stream>>>(bufA, src, dst, dis, bufB, E);
    // ---- layer 2: h2 = relu(bufB)@W2 -> bufA; bufD = b2 + dis^2*h2; scatter into bufD ----
    gemm_fused_kernel<true><<<n / 16, b256, 0, stream>>>(bufB, W2, b2, dis, bufA, bufD, 128, 128,
                                                         128);
    scatter128_kernel<<<(E + 7) / 8, b256, 0, stream>>>(bufA, src, dst, dis, bufD, E);
    // ---- layer 3: h3 = relu(bufD)@W3 -> bufC; d_out = b3 + dis^2*h3; scatter; softmax ----
    gemm_fused_kernel<true><<<n / 16, b256, 0, stream>>>(bufD, W3p, b3, dis, bufC, outp, 48, 40,
                                                         40);
    scatter40_kernel<<<(E + 7) / 8, b256, 0, stream>>>(bufC, src, dst, dis, outp, E);
    log_softmax_kernel<<<(n + 7) / 8, b256, 0, stream>>>(outp, n);
}